// VITModel_56435870270003
// MI455X (gfx1250) — compile-verified
//
#include <hip/hip_runtime.h>

// ---------------------------------------------------------------------------
// Problem constants (from reference): B=4, T=1024, S=1088, V=50257
// loss = -sum(valid * (feat[gt]-lse)) / max(sum(valid),1)
// eos_prob = mean over all B*S rows of exp(feat[EOS]-lse)
// ---------------------------------------------------------------------------
constexpr int B_ = 4;
constexpr int T_ = 1024;
constexpr int S_ = 1088;
constexpr int V_ = 50257;
constexpr int EOS_TOK = 50256;
constexpr int R_ = B_ * S_;            // 4352 rows, == 68 * 64 exactly
constexpr float L2E = 1.4426950408889634f;   // log2(e)

typedef __attribute__((ext_vector_type(2))) float v2f;
typedef __attribute__((ext_vector_type(8))) float v8f;

__device__ __forceinline__ float ex2(float x) { return __builtin_amdgcn_exp2f(x); }

// merge one scalar element into (m, s) online-logsumexp state
__device__ __forceinline__ void merge1(float& m, float& s, float x) {
  float nm = fmaxf(m, x);
  s = s * ex2((m - nm) * L2E) + ex2((x - nm) * L2E);
  m = nm;
}

// ---------------------------------------------------------------------------
// Stage 1: one 256-thread workgroup per row. Streams 50257 floats with
// float4 (b128) loads, online logsumexp with chunk-of-16 max (one rescale +
// 16 exp2 per 16 elements), LDS tree merge, then writes per-row partials.
// ---------------------------------------------------------------------------
__global__ __launch_bounds__(256) void vit_row_lse_kernel(
    const float* __restrict__ feature, const int* __restrict__ ids,
    const int* __restrict__ mask, float* __restrict__ ws_tok,
    float* __restrict__ ws_valid, float* __restrict__ ws_eos) {
  const int r = blockIdx.x;
  const int t = threadIdx.x;
  const float* __restrict__ row = feature + (size_t)r * (size_t)V_;

  // rows start at (r*V) dwords; V odd -> only dword-aligned. Peel to 16B.
  const int lead = (4 - (int)(((size_t)r * (size_t)V_) & 3)) & 3;
  const int n_rem = V_ - lead;
  const int N4 = n_rem >> 2;            // number of float4s in aligned middle
  const int tail = n_rem & 3;
  const float4* __restrict__ p4 = (const float4*)(row + lead);

  float m = -__builtin_huge_valf();
  float s = 0.0f;

  const int TPB = 256;
  const int U = 4;                      // float4s per thread per iteration
  const int iters = N4 / (TPB * U);     // full 16KB block tiles

  for (int it = 0; it < iters; ++it) {
    const int base = it * (TPB * U) + t;
    float4 a0 = p4[base];
    float4 a1 = p4[base + TPB];
    float4 a2 = p4[base + 2 * TPB];
    float4 a3 = p4[base + 3 * TPB];
    // stage next tile toward L2 (gfx1250 global_prefetch_b8)
    __builtin_prefetch((const void*)(p4 + base + 4 * TPB), 0, 1);

    float cm = fmaxf(
        fmaxf(fmaxf(fmaxf(a0.x, a0.y), fmaxf(a0.z, a0.w)),
              fmaxf(fmaxf(a1.x, a1.y), fmaxf(a1.z, a1.w))),
        fmaxf(fmaxf(fmaxf(a2.x, a2.y), fmaxf(a2.z, a2.w)),
              fmaxf(fmaxf(a3.x, a3.y), fmaxf(a3.z, a3.w))));
    float nm = fmaxf(m, cm);
    float s16 =
        ex2((a0.x - nm) * L2E) + ex2((a0.y - nm) * L2E) +
        ex2((a0.z - nm) * L2E) + ex2((a0.w - nm) * L2E) +
        ex2((a1.x - nm) * L2E) + ex2((a1.y - nm) * L2E) +
        ex2((a1.z - nm) * L2E) + ex2((a1.w - nm) * L2E) +
        ex2((a2.x - nm) * L2E) + ex2((a2.y - nm) * L2E) +
        ex2((a2.z - nm) * L2E) + ex2((a2.w - nm) * L2E) +
        ex2((a3.x - nm) * L2E) + ex2((a3.y - nm) * L2E) +
        ex2((a3.z - nm) * L2E) + ex2((a3.w - nm) * L2E);
    s = s * ex2((m - nm) * L2E) + s16;
    m = nm;
  }
  // remaining float4s (guarded)
  for (int i = iters * (TPB * U) + t; i < N4; i += TPB) {
    float4 a = p4[i];
    float cm = fmaxf(fmaxf(a.x, a.y), fmaxf(a.z, a.w));
    float nm = fmaxf(m, cm);
    float s4 = ex2((a.x - nm) * L2E) + ex2((a.y - nm) * L2E) +
               ex2((a.z - nm) * L2E) + ex2((a.w - nm) * L2E);
    s = s * ex2((m - nm) * L2E) + s4;
    m = nm;
  }
  // misaligned lead / tail scalars
  if (t < lead) merge1(m, s, row[t]);
  if (t < tail) merge1(m, s, row[lead + (N4 << 2) + t]);

  // block-level merge of (m, s) pairs
  __shared__ float sm[256];
  __shared__ float ss[256];
  sm[t] = m;
  ss[t] = s;
  __syncthreads();
  for (int w = 128; w > 0; w >>= 1) {
    if (t < w) {
      float m1 = sm[t], s1 = ss[t];
      float m2 = sm[t + w], s2 = ss[t + w];
      float nm = fmaxf(m1, m2);
      sm[t] = nm;
      ss[t] = s1 * ex2((m1 - nm) * L2E) + s2 * ex2((m2 - nm) * L2E);
    }
    __syncthreads();
  }

  if (t == 0) {
    const float lse = sm[0] + __logf(ss[0]);
    const int si = r % S_;
    const int b = r / S_;
    float tok = 0.0f;
    float validf = 0.0f;
    if (si + 1 < T_) {                       // shifted target exists & mask shifted
      const int mi = b * T_ + si + 1;
      if (mask[mi] != 0) {
        validf = 1.0f;
        const int gt = ids[mi];
        tok = row[gt] - lse;                 // log p(target)
      }
    }
    ws_tok[r] = tok;
    ws_valid[r] = validf;
    ws_eos[r] = ex2((row[EOS_TOK] - lse) * L2E);
  }
}

// ---------------------------------------------------------------------------
// Stage 2: single-wave deterministic reduction of the 3 x 4352 partials on
// the matrix pipe: V_WMMA_F32_16X16X4_F32 with A = ones gives C[i][j] +=
// sum_k B[k][j]; every row of D is the column-sum vector, so summing the
// first C VGPR over all 32 lanes yields 2x the grand total (rows M=0 and
// M=8), in full fp32. 4352 = 68 tiles of 64 values, no tail.
// ---------------------------------------------------------------------------
__global__ __launch_bounds__(32) void vit_final_reduce_kernel(
    const float* __restrict__ ws_tok, const float* __restrict__ ws_valid,
    const float* __restrict__ ws_eos, float* __restrict__ out) {
  const int l = threadIdx.x;
  v2f a;
  a.x = 1.0f;
  a.y = 1.0f;
  v8f ct = {};
  v8f cv = {};
  v8f ce = {};
  const int tiles = R_ >> 6;  // 68
  for (int tt = 0; tt < tiles; ++tt) {
    const int i0 = (tt << 6) + l;
    const int i1 = i0 + 32;
    v2f bt, bv, be;
    bt.x = ws_tok[i0];
    bt.y = ws_tok[i1];
    bv.x = ws_valid[i0];
    bv.y = ws_valid[i1];
    be.x = ws_eos[i0];
    be.y = ws_eos[i1];
    ct = __builtin_amdgcn_wmma_f32_16x16x4_f32(false, a, false, bt, (short)0,
                                               ct, false, false);
    cv = __builtin_amdgcn_wmma_f32_16x16x4_f32(false, a, false, bv, (short)0,
                                               cv, false, false);
    ce = __builtin_amdgcn_wmma_f32_16x16x4_f32(false, a, false, be, (short)0,
                                               ce, false, false);
  }
  float st = ct[0];
  float sv = cv[0];
  float se = ce[0];
  // 32-lane butterfly: all lanes end with 2x grand total
  for (int off = 16; off > 0; off >>= 1) {
    st += __shfl_xor(st, off, 32);
    sv += __shfl_xor(sv, off, 32);
    se += __shfl_xor(se, off, 32);
  }
  st *= 0.5f;
  sv *= 0.5f;
  se *= 0.5f;
  if (l == 0) {
    out[0] = -st / fmaxf(sv, 1.0f);          // mean CE over valid tokens
    out[1] = se * (1.0f / (float)R_);        // mean EOS probability
  }
}

// ---------------------------------------------------------------------------
extern "C" void kernel_launch(void* const* d_in, const int* in_sizes, int n_in,
                              void* d_out, int out_size, void* d_ws,
                              size_t ws_size, hipStream_t stream) {
  (void)in_sizes;
  (void)n_in;
  (void)out_size;
  (void)ws_size;
  const float* feature = (const float*)d_in[0];
  const int* ids = (const int*)d_in[1];
  const int* mask = (const int*)d_in[2];
  float* out = (float*)d_out;

  float* ws_tok = (float*)d_ws;        // R_ floats
  float* ws_valid = ws_tok + R_;       // R_ floats
  float* ws_eos = ws_valid + R_;       // R_ floats

  vit_row_lse_kernel<<<R_, 256, 0, stream>>>(feature, ids, mask, ws_tok,
                                             ws_valid, ws_eos);
  vit_final_reduce_kernel<<<1, 32, 0, stream>>>(ws_tok, ws_valid, ws_eos, out);
}